// WCE_J_SIMPL3D_71536975282966
// MI455X (gfx1250) — compile-verified
//
#include <hip/hip_runtime.h>

// ---------------------------------------------------------------------------
// WCE + Jaccard-style simplified 3D loss, MI455X (gfx1250, wave32).
//
// Roofline: 157 MB single-pass reads / 23.3 TB/s ~= 6.7 us floor; ~0.3 GFLOP
// total -> purely bandwidth bound. Hot loop = coalesced b128 NT loads +
// register accumulation. Deterministic 2-phase reduction via d_ws (no float
// atomics). Cross-lane wave sums use V_WMMA_F32_16X16X4_F32 (A = ones) as the
// CDNA5 reduction primitive: D[.][j] = B[0][j]+B[1][j]+B[2][j]+B[3][j]
//   with B.v0 = x, B.v1 = 0  ->  D[.][j] = x[j] + x[j+16]  (32 -> 16 fold),
// then 4 xor-shuffles finish 16 -> 1.
// ---------------------------------------------------------------------------

typedef float f32x2 __attribute__((ext_vector_type(2)));
typedef float f32x4 __attribute__((ext_vector_type(4)));
typedef float f32x8 __attribute__((ext_vector_type(8)));
typedef int   i32x4 __attribute__((ext_vector_type(4)));

static constexpr int   kC     = 4;
static constexpr int   kDHW   = 128 * 160 * 160;   // 3,276,800 (divisible by 4)
static constexpr int   kNVOX  = 2 * kDHW;          // 6,553,600 voxels
static constexpr int   kQ     = kNVOX / 4;         // float4 quads
static constexpr int   kNACC  = 21;                // 16 mat + 4 cnt + 1 ce
static constexpr int   kWSS   = 24;                // ws stride (floats) per block
static constexpr float kLAMB  = 0.5f;
static constexpr float kEPS   = 1e-8f;

// Full wave32 sum of x; result valid in lane 0 (and lanes 0..15).
__device__ __forceinline__ float wave_reduce_wmma(float x) {
  f32x2 a; a[0] = 1.0f; a[1] = 1.0f;   // A[i][k] = 1 for all i,k
  f32x2 b; b[0] = x;    b[1] = 0.0f;   // B[0][j],B[2][j] = x[j],x[j+16]; B[1],B[3]=0
  f32x8 c = {};
  f32x8 d = __builtin_amdgcn_wmma_f32_16x16x4_f32(
      /*neg_a=*/false, a, /*neg_b=*/false, b,
      /*c_mod=*/(short)0, c, /*reuse_a=*/false, /*reuse_b=*/false);
  float s = d[0];                      // lane l holds S[l & 15] = x[l&15] + x[(l&15)+16]
  s += __shfl_xor(s, 1);
  s += __shfl_xor(s, 2);
  s += __shfl_xor(s, 4);
  s += __shfl_xor(s, 8);               // lanes 0..15: sum of S[0..15] = full wave sum
  return s;
}

__global__ __launch_bounds__(256) void wce_phase1(
    const float* __restrict__ inp, const int* __restrict__ tgt,
    const float* __restrict__ wgt, float* __restrict__ ws) {
  float acc[kNACC];
#pragma unroll
  for (int a = 0; a < kNACC; ++a) acc[a] = 0.0f;

  const int tid    = threadIdx.x;
  const int gtid   = blockIdx.x * 256 + tid;
  const int stride = gridDim.x * 256;

  for (int q = gtid; q < kQ; q += stride) {
    const int v0   = q * 4;
    const int bidx = v0 / kDHW;               // batch index (0 or 1)
    const int s    = v0 - bidx * kDHW;
    const size_t cbase = (size_t)bidx * (size_t)(kC * kDHW) + (size_t)s;

    // 4 voxels x 4 channels: four coalesced 16B NT loads (channel planes are
    // 12.5 MB apart; contiguous along w within each plane).
    f32x4 x0 = __builtin_nontemporal_load((const f32x4*)(inp + cbase));
    f32x4 x1 = __builtin_nontemporal_load((const f32x4*)(inp + cbase + (size_t)kDHW));
    f32x4 x2 = __builtin_nontemporal_load((const f32x4*)(inp + cbase + (size_t)(2 * kDHW)));
    f32x4 x3 = __builtin_nontemporal_load((const f32x4*)(inp + cbase + (size_t)(3 * kDHW)));
    i32x4 tv = __builtin_nontemporal_load((const i32x4*)(tgt + v0));
    f32x4 wv = __builtin_nontemporal_load((const f32x4*)(wgt + v0));

#pragma unroll
    for (int k = 0; k < 4; ++k) {
      const float a0 = x0[k], a1 = x1[k], a2 = x2[k], a3 = x3[k];
      const float m  = fmaxf(fmaxf(a0, a1), fmaxf(a2, a3));
      const float e0 = __expf(a0 - m), e1 = __expf(a1 - m);
      const float e2 = __expf(a2 - m), e3 = __expf(a3 - m);
      const float sum = (e0 + e1) + (e2 + e3);
      const float inv = __builtin_amdgcn_rcpf(sum);
      const float p0 = e0 * inv, p1 = e1 * inv, p2 = e2 * inv, p3 = e3 * inv;

      int t = tv[k]; t = t < 0 ? 0 : (t > 3 ? 3 : t);   // clip(target, 0, c-1)
      const float f0 = (t == 0) ? 1.0f : 0.0f;
      const float f1 = (t == 1) ? 1.0f : 0.0f;
      const float f2 = (t == 2) ? 1.0f : 0.0f;
      const float f3 = (t == 3) ? 1.0f : 0.0f;
      const float xt = (t == 0) ? a0 : ((t == 1) ? a1 : ((t == 2) ? a2 : a3));

      acc[20] += wv[k] * ((xt - m) - __logf(sum));       // sum of w * log_p[t]
      acc[16] += f0; acc[17] += f1; acc[18] += f2; acc[19] += f3;

      acc[0]  += p0 * f0; acc[1]  += p0 * f1; acc[2]  += p0 * f2; acc[3]  += p0 * f3;
      acc[4]  += p1 * f0; acc[5]  += p1 * f1; acc[6]  += p1 * f2; acc[7]  += p1 * f3;
      acc[8]  += p2 * f0; acc[9]  += p2 * f1; acc[10] += p2 * f2; acc[11] += p2 * f3;
      acc[12] += p3 * f0; acc[13] += p3 * f1; acc[14] += p3 * f2; acc[15] += p3 * f3;
    }
  }

  // Block reduction: WMMA wave fold -> LDS -> 21 threads write block partials.
  __shared__ float partial[8][kNACC];
  const int lane = tid & 31;
  const int wv_i = tid >> 5;
#pragma unroll
  for (int a = 0; a < kNACC; ++a) {
    const float s = wave_reduce_wmma(acc[a]);
    if (lane == 0) partial[wv_i][a] = s;
  }
  __syncthreads();
  if (tid < kNACC) {
    float t = 0.0f;
#pragma unroll
    for (int w = 0; w < 8; ++w) t += partial[w][tid];
    ws[blockIdx.x * kWSS + tid] = t;
  }
}

__global__ __launch_bounds__(256) void wce_phase2(
    const float* __restrict__ ws, float* __restrict__ out, int nblk) {
  float acc[kNACC];
#pragma unroll
  for (int a = 0; a < kNACC; ++a) acc[a] = 0.0f;

  for (int b = threadIdx.x; b < nblk; b += 256) {
#pragma unroll
    for (int a = 0; a < kNACC; ++a) acc[a] += ws[b * kWSS + a];
  }

  __shared__ float partial[8][kNACC];
  const int lane = threadIdx.x & 31;
  const int wv_i = threadIdx.x >> 5;
#pragma unroll
  for (int a = 0; a < kNACC; ++a) {
    const float s = wave_reduce_wmma(acc[a]);
    if (lane == 0) partial[wv_i][a] = s;
  }
  __syncthreads();

  if (threadIdx.x == 0) {
    float tot[kNACC];
#pragma unroll
    for (int a = 0; a < kNACC; ++a) {
      float t = 0.0f;
#pragma unroll
      for (int w = 0; w < 8; ++w) t += partial[w][a];
      tot[a] = t;
    }
    // tot[0..15] = final_mat[i*4+j], tot[16..19] = class counts, tot[20] = ce sum
    float lossd = 0.0f;
#pragma unroll
    for (int i = 0; i < 4; ++i) {
#pragma unroll
      for (int j = 0; j < 4; ++j) {
        if (i == j) continue;
        const float ni = tot[16 + i], nj = tot[16 + j];
        if (ni * nj > 0.0f) {
          const float nis  = (ni > 0.0f) ? ni : 1.0f;
          const float njs  = (nj > 0.0f) ? nj : 1.0f;
          const float term = 0.5f * (tot[i * 4 + i] / nis - tot[i * 4 + j] / njs) + 0.5f;
          lossd += -kLAMB * logf(term + kEPS);           // POWER == 1
        }
      }
    }
    const float ce_loss = -(kLAMB / (float)kNVOX) * tot[20];
    out[0] = ce_loss + lossd;
  }
}

extern "C" void kernel_launch(void* const* d_in, const int* in_sizes, int n_in,
                              void* d_out, int out_size, void* d_ws, size_t ws_size,
                              hipStream_t stream) {
  const float* inp = (const float*)d_in[0];   // (2,4,128,160,160) fp32
  const int*   tgt = (const int*)d_in[1];     // (2,128,160,160)  int32
  const float* wgt = (const float*)d_in[2];   // (2,128,160,160)  fp32
  float* out = (float*)d_out;                 // scalar fp32
  float* ws  = (float*)d_ws;

  int nblk = 2048;                            // 16,384 wave32 in flight
  const size_t need = (size_t)nblk * kWSS * sizeof(float);
  if (ws_size < need) {
    nblk = (int)(ws_size / (kWSS * sizeof(float)));
    if (nblk < 1) nblk = 1;
  }

  wce_phase1<<<nblk, 256, 0, stream>>>(inp, tgt, wgt, ws);
  wce_phase2<<<1, 256, 0, stream>>>(ws, out, nblk);
}